// YOLOv2GridLoss_16269336118029
// MI455X (gfx1250) — compile-verified
//
#include <hip/hip_runtime.h>
#include <hip/hip_bf16.h>
#include <math.h>

// Problem constants (fixed by the reference setup)
#define BDIM 128
#define NBOX 50
#define HDIM 28
#define NANCH 5
#define NCLS 80
// channel stride within a (b,a) plane = H*W; plane stride = 85*H*W
#define CH_STRIDE 784
#define PLANE_STRIDE (85 * 784)       // 66640 floats
#define CELLS (BDIM * NANCH * HDIM * HDIM) // 501760
#define NREC (BDIM * NBOX)            // 6400

__device__ __forceinline__ float sigmoidf_(float x) {
  return 1.0f / (1.0f + __expf(-x));
}

// ---------------------------------------------------------------------------
// Phase B: sum of sigmoid(conf_logit)^2 over ALL cells (lambda_noobj term,
// over-counted at obj cells; phase C subtracts the correction).
// Confidence channel bytes are staged global -> LDS with the CDNA5 async
// load path (ASYNCcnt), then consumed from LDS.
// 490 blocks x 256 threads x 4 cells = 501760 cells exactly.
// ---------------------------------------------------------------------------
__global__ __launch_bounds__(256) void conf_noobj_kernel(
    const float* __restrict__ outp, float* __restrict__ partials) {
  __shared__ float data[1024];
  __shared__ float red[256];
  const unsigned t = threadIdx.x;
  const unsigned base_cell = blockIdx.x * 1024u;

#pragma unroll
  for (int k = 0; k < 4; ++k) {
    unsigned f = base_cell + (unsigned)k * 256u + t;     // flat cell id
    unsigned plane = f / 784u;                           // (b*A + a)
    unsigned s = f - plane * 784u;                       // j*28 + i
    unsigned off_bytes = (plane * (unsigned)PLANE_STRIDE + 4u * 784u + s) * 4u;
    unsigned lds_addr = (unsigned)(uintptr_t)(&data[k * 256 + t]);
    // CDNA5 async global->LDS load (GVS mode: SADDR base + 32-bit VGPR offset)
    asm volatile("global_load_async_to_lds_b32 %0, %1, %2"
                 :
                 : "v"(lds_addr), "v"(off_bytes), "s"(outp)
                 : "memory");
  }
  // Wait for this wave's async transfers (each thread reads back only the
  // LDS slots it loaded itself, so ASYNCcnt==0 is sufficient).
  asm volatile("s_wait_asynccnt 0" ::: "memory");

  float acc = 0.0f;
#pragma unroll
  for (int k = 0; k < 4; ++k) {
    float c = sigmoidf_(data[k * 256 + t]);
    acc += c * c;
  }
  red[t] = acc;
  __syncthreads();
  for (int w = 128; w > 0; w >>= 1) {
    if (t < (unsigned)w) red[t] += red[t + w];
    __syncthreads();
  }
  if (t == 0) partials[blockIdx.x] = 0.5f * red[0];  // LAMBDA_NOOBJ = 0.5
}

// ---------------------------------------------------------------------------
// Phase C: one thread per ground-truth record (b, n).
// Recomputes the target cell, dedupes (last write wins, matching scatter),
// gathers the 85 strided channel values at that cell, and emits the
// coord + obj-conf (+ noobj correction) + class-NLL contribution.
// ---------------------------------------------------------------------------
__global__ __launch_bounds__(256) void record_loss_kernel(
    const float* __restrict__ outp, const float* __restrict__ boxes,
    const int* __restrict__ cls, const float* __restrict__ anchors,
    float* __restrict__ rec_loss) {
  const int gid = blockIdx.x * 256 + threadIdx.x;   // 0..6399
  const int b = gid / NBOX;
  const int n = gid - b * NBOX;

  float aw[NANCH], ah[NANCH], aarea[NANCH];
#pragma unroll
  for (int a = 0; a < NANCH; ++a) {
    aw[a] = anchors[2 * a];
    ah[a] = anchors[2 * a + 1];
    aarea[a] = aw[a] * ah[a];
  }

  auto cell_of = [&](int m, int& gi, int& gj, int& bn,
                     float& gx, float& gy, float& gw, float& gh) {
    const float* bx = boxes + ((size_t)b * NBOX + m) * 4;
    float x = bx[0], y = bx[1], w = bx[2], h = bx[3];
    gx = x * 28.0f; gy = y * 28.0f; gw = w * 28.0f; gh = h * 28.0f;
    gi = (int)gx; gi = gi < 0 ? 0 : (gi > 27 ? 27 : gi);
    gj = (int)gy; gj = gj < 0 ? 0 : (gj > 27 ? 27 : gj);
    float gtw = w * 448.0f, gth = h * 448.0f;
    float garea = gtw * gth;
    float best = -1.0f; bn = 0;
#pragma unroll
    for (int a = 0; a < NANCH; ++a) {
      float iw = fminf(gtw, aw[a]);
      float ih = fminf(gth, ah[a]);
      float inter = iw * ih;
      float iou = inter / (garea + aarea[a] - inter);
      if (iou > best) { best = iou; bn = a; }   // first max wins (argmax)
    }
  };

  int gi, gj, bn; float gx, gy, gw, gh;
  cell_of(n, gi, gj, bn, gx, gy, gw, gh);

  // last-write-wins: a later record targeting the same cell overrides us
  bool winner = true;
  for (int m = n + 1; m < NBOX; ++m) {
    int gi2, gj2, bn2; float d0, d1, d2, d3;
    cell_of(m, gi2, gj2, bn2, d0, d1, d2, d3);
    if (gi2 == gi && gj2 == gj && bn2 == bn) { winner = false; break; }
  }
  if (!winner) { rec_loss[gid] = 0.0f; return; }

  const float* cell =
      outp + (size_t)(b * NANCH + bn) * PLANE_STRIDE + (gj * 28 + gi);
  float x0 = cell[0];
  float x1 = cell[1 * CH_STRIDE];
  float x2 = cell[2 * CH_STRIDE];
  float x3 = cell[3 * CH_STRIDE];
  float x4 = cell[4 * CH_STRIDE];

  float tx = gx - (float)gi, ty = gy - (float)gj;
  // scaled anchors = anchors / (448/28) = anchors / 16
  float tw = __logf(gw / (aw[bn] * 0.0625f) + 1e-16f);
  float th = __logf(gh / (ah[bn] * 0.0625f) + 1e-16f);

  float p0 = sigmoidf_(x0), p1 = sigmoidf_(x1), pc = sigmoidf_(x4);
  float e0 = p0 - tx, e1 = p1 - ty, e2 = x2 - tw, e3 = x3 - th;
  float loss = 5.0f * (e0 * e0 + e1 * e1 + e2 * e2 + e3 * e3); // LAMBDA_COORD
  // LAMBDA_OBJ*(conf-1)^2, minus the LAMBDA_NOOBJ*conf^2 phase B over-counted here
  loss += (pc - 1.0f) * (pc - 1.0f) - 0.5f * pc * pc;

  // class NLL via single-pass online log-softmax over the 80 strided logits
  int tc = cls[(size_t)b * NBOX + n];
  float mx = -3.4e38f, ssum = 0.0f, xt = 0.0f;
  for (int c = 0; c < NCLS; ++c) {
    float x = cell[(5 + c) * CH_STRIDE];
    if (c == tc) xt = x;
    if (x > mx) { ssum = ssum * __expf(mx - x) + 1.0f; mx = x; }
    else        { ssum += __expf(x - mx); }
  }
  loss += -(xt - mx - __logf(ssum));   // LAMBDA_CLASS = 1

  rec_loss[gid] = loss;
}

// ---------------------------------------------------------------------------
// Phase D: deterministic final reduction (fixed strided assignment + fixed
// tree -> bitwise identical across graph replays; no float atomics).
// ---------------------------------------------------------------------------
__global__ __launch_bounds__(256) void finalize_kernel(
    const float* __restrict__ partials, int npart,
    const float* __restrict__ recs, int nrec, float* __restrict__ outv) {
  __shared__ float red[256];
  const int t = threadIdx.x;
  float acc = 0.0f;
  for (int i = t; i < npart; i += 256) acc += partials[i];
  for (int i = t; i < nrec; i += 256) acc += recs[i];
  red[t] = acc;
  __syncthreads();
  for (int w = 128; w > 0; w >>= 1) {
    if (t < w) red[t] += red[t + w];
    __syncthreads();
  }
  if (t == 0) outv[0] = red[0] * (1.0f / (float)BDIM);
}

extern "C" void kernel_launch(void* const* d_in, const int* in_sizes, int n_in,
                              void* d_out, int out_size, void* d_ws, size_t ws_size,
                              hipStream_t stream) {
  (void)in_sizes; (void)n_in; (void)out_size; (void)ws_size;
  const float* outp    = (const float*)d_in[0];  // (128, 425, 28, 28) f32
  const float* boxes   = (const float*)d_in[1];  // (128, 50, 4) f32
  const int*   cls     = (const int*)d_in[2];    // (128, 50) i32
  const float* anchors = (const float*)d_in[3];  // (5, 2) f32

  float* ws       = (float*)d_ws;
  float* partials = ws;         // 490 floats
  float* recs     = ws + 512;   // 6400 floats (all fully rewritten each call)

  conf_noobj_kernel<<<CELLS / 1024, 256, 0, stream>>>(outp, partials);
  record_loss_kernel<<<NREC / 256, 256, 0, stream>>>(outp, boxes, cls, anchors, recs);
  finalize_kernel<<<1, 256, 0, stream>>>(partials, CELLS / 1024, recs, NREC,
                                         (float*)d_out);
}